// GATModel_65506841199115
// MI455X (gfx1250) — compile-verified
//
#include <hip/hip_runtime.h>
#include <math.h>

#define N_NODES   50000
#define N_EDGES   1600000
#define F_IN      256
#define HIDDEN    64
#define NHEADS    4
#define FDIM      256      // NHEADS * HIDDEN
#define NEG_SLOPE 0.2f

typedef __attribute__((ext_vector_type(2))) float v2f;
typedef __attribute__((ext_vector_type(8))) float v8f;

// ---------------------------------------------------------------------------
// GEMM: out[M, 256] = X[M, 256] @ W[256, 256], exact fp32 via WMMA 16x16x4.
// One wave per 16x16 output tile; 64 WMMA issues over K=256.
// A-layout (16x4 f32): lane L holds row L%16, K = 2*(L/16)+{0,1}  (float2 load)
// B-layout (4x16):     lane L holds col L%16, K = v + 2*(L/16) for VGPR v
// C/D-layout (16x16):  VGPR v, lane L -> M = v + 8*(L/16), N = L%16
// ---------------------------------------------------------------------------
__global__ __launch_bounds__(256) void gemm_wmma_f32(
    const float* __restrict__ X, const float* __restrict__ W,
    float* __restrict__ out, int M)
{
  const int lane = threadIdx.x & 31;
  const int wave = threadIdx.x >> 5;
  const int tile = blockIdx.x * 8 + wave;          // 16x16 tiles
  const int tilesPerRow = FDIM / 16;               // 16
  const int rowTile = tile / tilesPerRow;
  const int colTile = tile % tilesPerRow;
  if (rowTile * 16 >= M) return;                   // wave-uniform guard

  const int half = lane >> 4;                      // 0 or 1
  const int l16  = lane & 15;
  const int row  = rowTile * 16 + l16;             // A row for this lane
  const int col  = colTile * 16 + l16;             // B/C column for this lane

  v8f acc = {};
  for (int k0 = 0; k0 < F_IN; k0 += 4) {
    const int k = k0 + 2 * half;
    v2f a;
    a.x = X[(size_t)row * F_IN + k];
    a.y = X[(size_t)row * F_IN + k + 1];
    v2f b;
    b.x = W[(size_t)(k)     * FDIM + col];
    b.y = W[(size_t)(k + 1) * FDIM + col];
    acc = __builtin_amdgcn_wmma_f32_16x16x4_f32(
        /*neg_a=*/false, a, /*neg_b=*/false, b,
        /*c_mod=*/(short)0, acc, /*reuse_a=*/false, /*reuse_b=*/false);
  }

  const int rbase = rowTile * 16 + 8 * half;
#pragma unroll
  for (int v = 0; v < 8; ++v)
    out[(size_t)(rbase + v) * FDIM + col] = acc[v];
}

// ---------------------------------------------------------------------------
__global__ __launch_bounds__(256) void fill_kernel(float* __restrict__ p, float v, int n)
{
  int i = blockIdx.x * blockDim.x + threadIdx.x;
  if (i < n) p[i] = v;
}

// el[n,h] = <feat[n,h,:], al[h,:]>, er likewise (feat read once for both).
__global__ __launch_bounds__(256) void el_er_kernel(
    const float* __restrict__ feat, const float* __restrict__ al,
    const float* __restrict__ ar, float* __restrict__ el, float* __restrict__ er)
{
  int i = blockIdx.x * blockDim.x + threadIdx.x;   // n*NHEADS + h
  if (i >= N_NODES * NHEADS) return;
  int n = i / NHEADS, h = i % NHEADS;
  const float* f = feat + (size_t)n * FDIM + h * HIDDEN;
  const float* a = al + h * HIDDEN;
  const float* b = ar + h * HIDDEN;
  float sl = 0.f, sr = 0.f;
#pragma unroll 8
  for (int d = 0; d < HIDDEN; ++d) {
    float fv = f[d];
    sl += fv * a[d];
    sr += fv * b[d];
  }
  el[i] = sl;
  er[i] = sr;
}

// Float atomic-max via signed-int max (vals >= 0) / unsigned-int min (vals < 0).
__device__ inline void atomicMaxFloat(float* addr, float val)
{
  if (val >= 0.f) atomicMax((int*)addr, __float_as_int(val));
  else            atomicMin((unsigned int*)addr, __float_as_uint(val));
}

// Pass 1: e = LeakyReLU(el[src] + er[dst]); segment-max into emax[dst].
__global__ __launch_bounds__(256) void edge_pass1(
    const int* __restrict__ src, const int* __restrict__ dst,
    const float* __restrict__ el, const float* __restrict__ er,
    float* __restrict__ ebuf, float* __restrict__ emax)
{
  int e = blockIdx.x * blockDim.x + threadIdx.x;
  if (e >= N_EDGES) return;
  int s = src[e], d = dst[e];
#pragma unroll
  for (int h = 0; h < NHEADS; ++h) {
    float v = el[s * NHEADS + h] + er[d * NHEADS + h];
    v = v > 0.f ? v : NEG_SLOPE * v;
    ebuf[(size_t)e * NHEADS + h] = v;
    atomicMaxFloat(&emax[d * NHEADS + h], v);
  }
}

// Pass 2: ex = exp(e - emax[dst]); segment-sum into denom[dst].
__global__ __launch_bounds__(256) void edge_pass2(
    const int* __restrict__ dst, float* __restrict__ ebuf,
    const float* __restrict__ emax, float* __restrict__ denom)
{
  int e = blockIdx.x * blockDim.x + threadIdx.x;
  if (e >= N_EDGES) return;
  int d = dst[e];
#pragma unroll
  for (int h = 0; h < NHEADS; ++h) {
    float ex = __expf(ebuf[(size_t)e * NHEADS + h] - emax[d * NHEADS + h]);
    ebuf[(size_t)e * NHEADS + h] = ex;
    atomicAdd(&denom[d * NHEADS + h], ex);
  }
}

// Pass 3: accum[dst] += alpha * feat[src]. 64 threads/edge, float4 per thread.
// feat (51MB) + accum (51MB) are L2-resident on MI455X (192MB L2).
__global__ __launch_bounds__(256) void edge_pass3(
    const int* __restrict__ src, const int* __restrict__ dst,
    const float* __restrict__ ebuf, const float* __restrict__ denom,
    const float* __restrict__ feat, float* __restrict__ accum)
{
  long long t = (long long)blockIdx.x * blockDim.x + threadIdx.x;
  if (t >= (long long)N_EDGES * 64) return;
  int e   = (int)(t >> 6);
  int sub = (int)(t & 63);
  int h   = sub >> 4;
  int d4  = (sub & 15) * 4;
  int s = src[e], dn = dst[e];
  float den = denom[dn * NHEADS + h];
  float alpha = ebuf[(size_t)e * NHEADS + h] / (den > 0.f ? den : 1.f);
  const float4 f = *(const float4*)(feat + (size_t)s * FDIM + h * HIDDEN + d4);
  float* o = accum + (size_t)dn * FDIM + h * HIDDEN + d4;
  atomicAdd(o + 0, alpha * f.x);
  atomicAdd(o + 1, alpha * f.y);
  atomicAdd(o + 2, alpha * f.z);
  atomicAdd(o + 3, alpha * f.w);
}

// Finalize: out = ELU(acc (+ residual) + bias)
__global__ __launch_bounds__(256) void finalize_kernel(
    const float* __restrict__ acc, const float* __restrict__ bias,
    const float* __restrict__ residual, float* __restrict__ out)
{
  int i = blockIdx.x * blockDim.x + threadIdx.x;
  if (i >= N_NODES * FDIM) return;
  int c = i & (FDIM - 1);
  float v = acc[i] + bias[c];
  if (residual) v += residual[i];
  out[i] = v > 0.f ? v : (__expf(v) - 1.f);
}

// ---------------------------------------------------------------------------
extern "C" void kernel_launch(void* const* d_in, const int* in_sizes, int n_in,
                              void* d_out, int out_size, void* d_ws, size_t ws_size,
                              hipStream_t stream)
{
  const float* inputs = (const float*)d_in[0];
  const int*   src    = (const int*)  d_in[1];
  const int*   dst    = (const int*)  d_in[2];
  const float* W1     = (const float*)d_in[3];
  const float* al1    = (const float*)d_in[4];
  const float* ar1    = (const float*)d_in[5];
  const float* b1     = (const float*)d_in[6];
  const float* W2     = (const float*)d_in[7];
  const float* al2    = (const float*)d_in[8];
  const float* ar2    = (const float*)d_in[9];
  const float* b2     = (const float*)d_in[10];
  float* out = (float*)d_out;

  // Workspace carve (floats): featA | hbuf | ebuf | el | er | emax | denom
  float* featA = (float*)d_ws;                                  // N*256
  float* hbuf  = featA + (size_t)N_NODES * FDIM;                // N*256
  float* ebuf  = hbuf  + (size_t)N_NODES * FDIM;                // E*4
  float* el    = ebuf  + (size_t)N_EDGES * NHEADS;              // N*4
  float* er    = el    + (size_t)N_NODES * NHEADS;
  float* emax  = er    + (size_t)N_NODES * NHEADS;
  float* denom = emax  + (size_t)N_NODES * NHEADS;

  const int BT = 256;
  const int gemmBlocks  = ((N_NODES / 16) * (FDIM / 16) + 7) / 8;  // 8 waves/block
  const int nhBlocks    = (N_NODES * NHEADS + BT - 1) / BT;
  const int edgeBlocks  = (N_EDGES + BT - 1) / BT;
  const long long p3    = (long long)N_EDGES * 64;
  const int p3Blocks    = (int)((p3 + BT - 1) / BT);
  const int featBlocks  = (N_NODES * FDIM + BT - 1) / BT;

  // ---------------- Layer 1 ----------------
  gemm_wmma_f32<<<gemmBlocks, BT, 0, stream>>>(inputs, W1, featA, N_NODES);
  el_er_kernel <<<nhBlocks,  BT, 0, stream>>>(featA, al1, ar1, el, er);
  fill_kernel  <<<nhBlocks,  BT, 0, stream>>>(emax, -INFINITY, N_NODES * NHEADS);
  fill_kernel  <<<nhBlocks,  BT, 0, stream>>>(denom, 0.f,      N_NODES * NHEADS);
  fill_kernel  <<<featBlocks,BT, 0, stream>>>(hbuf, 0.f,       N_NODES * FDIM);
  edge_pass1   <<<edgeBlocks,BT, 0, stream>>>(src, dst, el, er, ebuf, emax);
  edge_pass2   <<<edgeBlocks,BT, 0, stream>>>(dst, ebuf, emax, denom);
  edge_pass3   <<<p3Blocks,  BT, 0, stream>>>(src, dst, ebuf, denom, featA, hbuf);
  finalize_kernel<<<featBlocks, BT, 0, stream>>>(hbuf, b1, nullptr, hbuf);

  // ---------------- Layer 2 (residual) ----------------
  gemm_wmma_f32<<<gemmBlocks, BT, 0, stream>>>(hbuf, W2, featA, N_NODES);
  el_er_kernel <<<nhBlocks,  BT, 0, stream>>>(featA, al2, ar2, el, er);
  fill_kernel  <<<nhBlocks,  BT, 0, stream>>>(emax, -INFINITY, N_NODES * NHEADS);
  fill_kernel  <<<nhBlocks,  BT, 0, stream>>>(denom, 0.f,      N_NODES * NHEADS);
  fill_kernel  <<<featBlocks,BT, 0, stream>>>(out, 0.f,        N_NODES * FDIM);
  edge_pass1   <<<edgeBlocks,BT, 0, stream>>>(src, dst, el, er, ebuf, emax);
  edge_pass2   <<<edgeBlocks,BT, 0, stream>>>(dst, ebuf, emax, denom);
  edge_pass3   <<<p3Blocks,  BT, 0, stream>>>(src, dst, ebuf, denom, featA, out);
  finalize_kernel<<<featBlocks, BT, 0, stream>>>(out, b2, hbuf, out);
}